// CompactBilinearLayer_12549894439328
// MI455X (gfx1250) — compile-verified
//
#include <hip/hip_runtime.h>

// ---------------------------------------------------------------------------
// Compact Bilinear Pooling on MI455X (gfx1250, wave32, WMMA).
//
//   y[b] = l2norm( signed_sqrt( sum_hw ifft( fft(sketch1(x)) * fft(sketch2(x)) ) ) )
//
// Design:
//  * sketch = scatter (count-sketch matrices are 1-sparse per row)
//  * z = p1 + i*p2  -> single complex FFT-8192 per pixel (pack trick)
//  * FFT-8192 = four-step 64x128, both stages as complex f32 matmuls on
//    V_WMMA_F32_16X16X4_F32. DFT factors come from tiny LDS v2f tables
//    (64 + 128 entries, bank-conflict-free); hot loops use explicit
//    address_space(3) pointers so every access is native 32-bit LDS
//    addressing (no generic-pointer null checks).
//  * Hermitian spectral product accumulated per batch (4097 bins), then one
//    inverse WMMA-FFT per batch + signed-sqrt + L2 normalize.
// ---------------------------------------------------------------------------

typedef __attribute__((ext_vector_type(2))) float v2f;
typedef __attribute__((ext_vector_type(8))) float v8f;

// LDS-qualified element types (pointers to these use 32-bit DS addressing).
typedef __attribute__((address_space(3))) float lds_float;
typedef __attribute__((address_space(3))) v2f   lds_v2f;

static __device__ __forceinline__ lds_float* to_lds(float* p) {
  return (lds_float*)p;
}
static __device__ __forceinline__ const lds_v2f* to_lds(const v2f* p) {
  return (const lds_v2f*)p;
}

#define TWO_PI_F 6.28318530717958647692f

static __device__ __forceinline__ v8f wmma_f32_k4(v2f a, v2f b, v8f c) {
  // D(16x16,f32) = A(16x4,f32) * B(4x16,f32) + C
  return __builtin_amdgcn_wmma_f32_16x16x4_f32(
      /*neg_a=*/false, a, /*neg_b=*/false, b,
      /*c_mod=*/(short)0, c, /*reuse_a=*/false, /*reuse_b=*/false);
}

// Build unit-root tables: tN[j] = (cos(2*pi*j/N), sin(2*pi*j/N)).
// Call with all 256 threads, then barrier at the call site.
static __device__ __forceinline__ void init_twiddle_tables(v2f* t64,
                                                           v2f* t128,
                                                           int tid) {
  for (int j = tid; j < 64; j += 256) {
    float s, c;
    __sincosf(TWO_PI_F * (float)j * (1.0f / 64.0f), &s, &c);
    v2f t; t[0] = c; t[1] = s;
    t64[j] = t;
  }
  for (int j = tid; j < 128; j += 256) {
    float s, c;
    __sincosf(TWO_PI_F * (float)j * (1.0f / 128.0f), &s, &c);
    v2f t; t[0] = c; t[1] = s;
    t128[j] = t;
  }
}

// In-place complex FFT of length 8192 on LDS arrays zr/zi.
// Input  : natural order, z[n] at index n, with n = n1*128 + n2.
// Output : X[k2*64 + k1] stored at index k1*128 + k2
//          (i.e. X[k] lives at ((k & 63) << 7) | (k >> 6)).
// dir = -1.0f forward DFT, +1.0f inverse DFT (unscaled).
// Must be called by all 256 threads (8 wave32s), uniform control flow.
static __device__ __forceinline__ void fft8192_wmma(
    lds_float* zr, lds_float* zi,
    const lds_v2f* t64, const lds_v2f* t128,
    float dir, int tid) {
  const int lane = tid & 31;
  const int wv   = tid >> 5;     // wave id 0..7 -> column-tile tn
  const int mloc = lane & 15;
  const int kp   = lane >> 4;    // K half-select (0/1)
  const int col  = wv * 16 + mloc;

  const v8f VZERO = {};
  v8f cr[4], ci[4];

  // ---------------- Step 1: C1[k1][n2] = sum_n1 D64[k1][n1] * Z[n1][n2] ----
  // M=64 (k1), N=128 (n2), K=64 (n1). Wave 'wv' owns columns [wv*16, wv*16+16).
#pragma unroll
  for (int t = 0; t < 4; ++t) { cr[t] = VZERO; ci[t] = VZERO; }

  for (int kc = 0; kc < 16; ++kc) {
    const int r0 = kc * 4 + 2 * kp;           // n1 of vector element v: r0+v
    v2f br, bi;                               // B frag: Z[n1][col]
    br[0] = zr[(r0    ) * 128 + col];
    br[1] = zr[(r0 + 1) * 128 + col];
    bi[0] = zi[(r0    ) * 128 + col];
    bi[1] = zi[(r0 + 1) * 128 + col];
#pragma unroll
    for (int tm = 0; tm < 4; ++tm) {
      const int k1 = tm * 16 + mloc;          // A frag row
      v2f ar, ai, ain;
#pragma unroll
      for (int v = 0; v < 2; ++v) {
        const int n1 = r0 + v;
        const v2f t = t64[(k1 * n1) & 63];    // e^{i*2pi*idx/64}
        const float si = dir * t[1];
        ar[v] = t[0]; ai[v] = si; ain[v] = -si;
      }
      // complex MAC: (Cr,Ci) += (Ar+iAi)(Br+iBi)
      cr[tm] = wmma_f32_k4(ar,  br, cr[tm]);
      cr[tm] = wmma_f32_k4(ain, bi, cr[tm]);
      ci[tm] = wmma_f32_k4(ar,  bi, ci[tm]);
      ci[tm] = wmma_f32_k4(ai,  br, ci[tm]);
    }
  }

  // Twiddle: *= W8192^(k1*n2)  (only 32 sincos per lane per FFT)
#pragma unroll
  for (int tm = 0; tm < 4; ++tm) {
#pragma unroll
    for (int v = 0; v < 8; ++v) {
      const int k1 = tm * 16 + v + 8 * kp;    // C/D layout: M = v + 8*kp
      const float ang =
          dir * TWO_PI_F * (float)((k1 * col) & 8191) * (1.0f / 8192.0f);
      float s, c;
      __sincosf(ang, &s, &c);
      const float r = cr[tm][v], im = ci[tm][v];
      cr[tm][v] = r * c - im * s;
      ci[tm][v] = r * s + im * c;
    }
  }

  __syncthreads();                            // all reads of Z complete
#pragma unroll
  for (int tm = 0; tm < 4; ++tm)
#pragma unroll
    for (int v = 0; v < 8; ++v) {
      const int row = tm * 16 + v + 8 * kp;
      zr[row * 128 + col] = cr[tm][v];
      zi[row * 128 + col] = ci[tm][v];
    }
  __syncthreads();

  // ---------------- Step 3: C2[k1][k2] = sum_n2 C1[k1][n2] * D128[n2][k2] --
  // M=64 (k1), N=128 (k2), K=128 (n2).
#pragma unroll
  for (int t = 0; t < 4; ++t) { cr[t] = VZERO; ci[t] = VZERO; }

  for (int kc = 0; kc < 32; ++kc) {
    const int r0 = kc * 4 + 2 * kp;           // n2 of vector element v: r0+v
    v2f br, bi;                               // B frag: D128[n2][col]
#pragma unroll
    for (int v = 0; v < 2; ++v) {
      const int n2 = r0 + v;
      const v2f t = t128[(n2 * col) & 127];
      br[v] = t[0]; bi[v] = dir * t[1];
    }
#pragma unroll
    for (int tm = 0; tm < 4; ++tm) {
      const int row = tm * 16 + mloc;         // k1
      v2f ar, ai, ain;
      ar[0] = zr[row * 128 + r0];
      ar[1] = zr[row * 128 + r0 + 1];
      ai[0] = zi[row * 128 + r0];
      ai[1] = zi[row * 128 + r0 + 1];
      ain[0] = -ai[0]; ain[1] = -ai[1];
      cr[tm] = wmma_f32_k4(ar,  br, cr[tm]);
      cr[tm] = wmma_f32_k4(ain, bi, cr[tm]);
      ci[tm] = wmma_f32_k4(ar,  bi, ci[tm]);
      ci[tm] = wmma_f32_k4(ai,  br, ci[tm]);
    }
  }

  __syncthreads();
#pragma unroll
  for (int tm = 0; tm < 4; ++tm)
#pragma unroll
    for (int v = 0; v < 8; ++v) {
      const int row = tm * 16 + v + 8 * kp;
      zr[row * 128 + col] = cr[tm][v];
      zi[row * 128 + col] = ci[tm][v];
    }
  __syncthreads();
}

// Map natural spectral index k -> LDS address after fft8192_wmma.
static __device__ __forceinline__ int spec_addr(int k) {
  return ((k & 63) << 7) | (k >> 6);
}

// ---------------------------------------------------------------------------
// Kernel 1: recover count-sketch (h, s) from the dense M1/M2 (1-sparse rows).
// ---------------------------------------------------------------------------
__global__ void cbp_extract_sketch(const float* __restrict__ M1,
                                   const float* __restrict__ M2,
                                   int* __restrict__ h1, float* __restrict__ s1,
                                   int* __restrict__ h2, float* __restrict__ s2) {
  const int t = blockIdx.x * blockDim.x + threadIdx.x;
  if (t >= 1024) return;
  const int m = t >> 9;             // which matrix
  const int c = t & 511;            // input channel
  const float* row = (m ? M2 : M1) + (size_t)c * 8192;
  int h = 0;
  float s = 0.0f;
  for (int j = 0; j < 8192; ++j) {
    const float v = row[j];
    if (v != 0.0f) { h = j; s = v; }
  }
  if (m) { h2[c] = h; s2[c] = s; }
  else   { h1[c] = h; s1[c] = s; }
}

// ---------------------------------------------------------------------------
// Kernel 2: per 7-pixel group: sketch-scatter -> forward WMMA FFT ->
// Hermitian spectral product -> LDS accumulate -> global atomic accumulate.
// grid = (28 groups, 32 batches), block = 256 (8 wave32s).
// ---------------------------------------------------------------------------
__global__ __launch_bounds__(256)
void cbp_fft_accum(const float* __restrict__ x,
                   const int* __restrict__ h1, const int* __restrict__ h2,
                   const float* __restrict__ s1, const float* __restrict__ s2,
                   float* __restrict__ accR, float* __restrict__ accI) {
  __shared__ float zr[8192];
  __shared__ float zi[8192];
  __shared__ float sR[4104];   // Hermitian half of spectral-product sum
  __shared__ float sI[4104];
  __shared__ v2f t64[64];
  __shared__ v2f t128[128];

  lds_float* lzr = to_lds(zr);
  lds_float* lzi = to_lds(zi);
  const lds_v2f* lt64  = to_lds((const v2f*)t64);
  const lds_v2f* lt128 = to_lds((const v2f*)t128);

  const int tid = threadIdx.x;
  const int g   = blockIdx.x;  // 0..27 spatial group
  const int b   = blockIdx.y;  // 0..31 batch

  init_twiddle_tables(t64, t128, tid);
  for (int k = tid; k < 4104; k += 256) { sR[k] = 0.0f; sI[k] = 0.0f; }

  for (int p = 0; p < 7; ++p) {
    const int n = b * 196 + g * 7 + p;   // pixel row in x_flat

    // z = p1 + i*p2 via count-sketch scatter
    for (int i = tid; i < 8192; i += 256) { zr[i] = 0.0f; zi[i] = 0.0f; }
    __syncthreads();
    for (int c = tid; c < 512; c += 256) {
      const float xv = x[(size_t)n * 512 + c];
      atomicAdd(&zr[h1[c]], s1[c] * xv);
      atomicAdd(&zi[h2[c]], s2[c] * xv);
    }
    __syncthreads();

    fft8192_wmma(lzr, lzi, lt64, lt128, -1.0f, tid);

    // F1 = (Z[k] + Z*[N-k]) / 2 ; F2 = (Z[k] - Z*[N-k]) / (2i) ; S += F1*F2
    for (int k = tid; k <= 4096; k += 256) {
      const int kn = (8192 - k) & 8191;
      const int ak = spec_addr(k);
      const int an = spec_addr(kn);
      const float a  = zr[ak], bb = zi[ak];
      const float c  = zr[an], d  = zi[an];
      const float f1r = 0.5f * (a + c);
      const float f1i = 0.5f * (bb - d);
      const float f2r = 0.5f * (bb + d);
      const float f2i = -0.5f * (a - c);
      sR[k] += f1r * f2r - f1i * f2i;
      sI[k] += f1r * f2i + f1i * f2r;
    }
    __syncthreads();   // z consumed; safe to rebuild next pixel
  }

  for (int k = tid; k <= 4096; k += 256) {
    atomicAdd(&accR[b * 4097 + k], sR[k]);
    atomicAdd(&accI[b * 4097 + k], sI[k]);
  }
}

// ---------------------------------------------------------------------------
// Kernel 3: per batch: rebuild full Hermitian spectrum, inverse WMMA FFT,
// signed sqrt, L2 normalize, write output. grid = 32, block = 256.
// ---------------------------------------------------------------------------
__global__ __launch_bounds__(256)
void cbp_ifft_finalize(const float* __restrict__ accR,
                       const float* __restrict__ accI,
                       float* __restrict__ out) {
  __shared__ float zr[8192];
  __shared__ float zi[8192];
  __shared__ float red[256];
  __shared__ v2f t64[64];
  __shared__ v2f t128[128];

  lds_float* lzr = to_lds(zr);
  lds_float* lzi = to_lds(zi);
  const lds_v2f* lt64  = to_lds((const v2f*)t64);
  const lds_v2f* lt128 = to_lds((const v2f*)t128);

  const int tid = threadIdx.x;
  const int b   = blockIdx.x;

  init_twiddle_tables(t64, t128, tid);
  for (int j = tid; j < 8192; j += 256) {
    float r, im;
    if (j <= 4096) {
      r  = accR[b * 4097 + j];
      im = accI[b * 4097 + j];
    } else {
      r  =  accR[b * 4097 + (8192 - j)];
      im = -accI[b * 4097 + (8192 - j)];
    }
    zr[j] = r;
    zi[j] = im;
  }
  __syncthreads();

  fft8192_wmma(lzr, lzi, lt64, lt128, +1.0f, tid);   // unscaled inverse DFT

  // sum |v| == sum signed_sqrt(v)^2
  float ss = 0.0f;
  for (int j = tid; j < 8192; j += 256)
    ss += fabsf(zr[spec_addr(j)]) * (1.0f / 8192.0f);
  red[tid] = ss;
  __syncthreads();
  for (int off = 128; off > 0; off >>= 1) {
    if (tid < off) red[tid] += red[tid + off];
    __syncthreads();
  }
  const float inv = rsqrtf(fmaxf(red[0], 1e-10f));

  for (int j = tid; j < 8192; j += 256) {
    const float v = zr[spec_addr(j)] * (1.0f / 8192.0f);
    out[(size_t)b * 8192 + j] = copysignf(sqrtf(fabsf(v)), v) * inv;
  }
}

// ---------------------------------------------------------------------------
extern "C" void kernel_launch(void* const* d_in, const int* in_sizes, int n_in,
                              void* d_out, int out_size, void* d_ws,
                              size_t ws_size, hipStream_t stream) {
  (void)in_sizes; (void)n_in; (void)out_size; (void)ws_size;
  const float* x  = (const float*)d_in[0];
  const float* M1 = (const float*)d_in[1];
  const float* M2 = (const float*)d_in[2];

  char* ws = (char*)d_ws;
  int*   h1   = (int*)  (ws + 0);
  int*   h2   = (int*)  (ws + 2048);
  float* s1   = (float*)(ws + 4096);
  float* s2   = (float*)(ws + 6144);
  float* accR = (float*)(ws + 8192);
  float* accI = accR + 32 * 4097;

  // Zero batch spectral accumulators every launch (capture-safe memset node).
  hipMemsetAsync(accR, 0, (size_t)32 * 4097 * 2 * sizeof(float), stream);

  cbp_extract_sketch<<<2, 512, 0, stream>>>(M1, M2, h1, s1, h2, s2);

  dim3 grid_fft(28, 32);
  cbp_fft_accum<<<grid_fft, 256, 0, stream>>>(x, h1, h2, s1, s2, accR, accI);

  cbp_ifft_finalize<<<32, 256, 0, stream>>>(accR, accI, (float*)d_out);
}